// SAGE_67551245631643
// MI455X (gfx1250) — compile-verified
//
#include <hip/hip_runtime.h>
#include <hip/hip_bf16.h>

typedef float v2f __attribute__((ext_vector_type(2)));
typedef float v8f __attribute__((ext_vector_type(8)));

#define N_NODES 50000
#define N_EDGES 800000

// ---------------------------------------------------------------------------
// Degree counting: deg2[row] (out-degree, for Laplacian), deg1[col] (in-degree,
// self-loop added later in dinv kernel)
// ---------------------------------------------------------------------------
__global__ void deg_kernel(const int* __restrict__ edges, float* __restrict__ deg1,
                           float* __restrict__ deg2, int E) {
    int e = blockIdx.x * blockDim.x + threadIdx.x;
    if (e >= E) return;
    int r = edges[e];
    int c = edges[E + e];
    atomicAdd(&deg2[r], 1.0f);
    atomicAdd(&deg1[c], 1.0f);
}

__global__ void dinv_kernel(const float* __restrict__ deg1, const float* __restrict__ deg2,
                            float* __restrict__ dinv1, float* __restrict__ dinv2, int N) {
    int i = blockIdx.x * blockDim.x + threadIdx.x;
    if (i >= N) return;
    dinv1[i] = rsqrtf(deg1[i] + 1.0f);           // +1 = self loop; always > 0
    float d2 = deg2[i];
    dinv2[i] = (d2 > 0.0f) ? rsqrtf(d2) : 0.0f;
}

// ---------------------------------------------------------------------------
// FP32 WMMA GEMM: C[M,N] = act(A[M,K] @ B[K,N] + bias). One wave per 16x16
// tile, V_WMMA_F32_16X16X4_F32, K stepped by 4. M,N multiples of 16; K of 4.
// ACT: 0 = identity, 1 = tanh.
// ---------------------------------------------------------------------------
template <int ACT>
__global__ void gemm_wmma_f32(const float* __restrict__ A, const float* __restrict__ B,
                              const float* __restrict__ bias, float* __restrict__ C,
                              int M, int N, int K) {
    int wave = (blockIdx.x * blockDim.x + threadIdx.x) >> 5;
    int lane = threadIdx.x & 31;
    int ntiles = N >> 4;
    int tm = wave / ntiles;
    int tn = wave - tm * ntiles;
    if (tm * 16 >= M) return;                    // wave-uniform: EXEC stays all-ones

    const int m0 = tm << 4, n0 = tn << 4;
    const int mrow = m0 + (lane & 15);           // A row held by this lane
    const int ncol = n0 + (lane & 15);           // B col held by this lane
    const int kh   = (lane >> 4) << 1;           // K sub-offset: 0 (lanes 0-15) or 2

    v8f acc = {};
    const float* Ap = A + (size_t)mrow * K;
#pragma unroll 8
    for (int k = 0; k < K; k += 4) {
        v2f a, b;
        a.x = Ap[k + kh];
        a.y = Ap[k + kh + 1];
        b.x = B[(size_t)(k + kh) * N + ncol];
        b.y = B[(size_t)(k + kh + 1) * N + ncol];
        acc = __builtin_amdgcn_wmma_f32_16x16x4_f32(false, a, false, b,
                                                    (short)0, acc, false, false);
    }

    float bn = bias ? bias[ncol] : 0.0f;
#pragma unroll
    for (int v = 0; v < 8; ++v) {
        int rrow = m0 + v + ((lane >> 4) << 3);  // C/D layout: M = v (+8 for hi half)
        float val = acc[v] + bn;
        if (ACT == 1) val = tanhf(val);
        C[(size_t)rrow * N + ncol] = val;
    }
}

// ---------------------------------------------------------------------------
// H init: self-loop + bias.  H[i,d] = b1[d] + dinv1[i]^2 * XW[i,d]
// ---------------------------------------------------------------------------
__global__ void h_init_kernel(const float* __restrict__ XW, const float* __restrict__ b1,
                              const float* __restrict__ dinv1, float* __restrict__ H) {
    int i = blockIdx.x;
    int d = threadIdx.x;                         // 256 = D1
    float di = dinv1[i];
    H[(size_t)i * 256 + d] = b1[d] + di * di * XW[(size_t)i * 256 + d];
}

// Edge scatter: H[col,d] += dinv1[row]*dinv1[col] * XW[row,d]
__global__ void h_scatter_kernel(const int* __restrict__ edges, const float* __restrict__ XW,
                                 const float* __restrict__ dinv1, float* __restrict__ H, int E) {
    int e = blockIdx.x;
    int d = threadIdx.x;                         // 256
    int j = edges[e];                            // source
    int i = edges[E + e];                        // target
    float w = dinv1[i] * dinv1[j];
    atomicAdd(&H[(size_t)i * 256 + d], w * XW[(size_t)j * 256 + d]);
}

// ---------------------------------------------------------------------------
// Row softmax over K=32 (one wave per row, in place) + row sums of S
// ---------------------------------------------------------------------------
__global__ void softmax_kernel(float* __restrict__ S, float* __restrict__ rs, int N) {
    int row  = (blockIdx.x * blockDim.x + threadIdx.x) >> 5;
    int lane = threadIdx.x & 31;
    if (row >= N) return;
    float v = S[(size_t)row * 32 + lane];
    float m = v;
    for (int o = 16; o > 0; o >>= 1) m = fmaxf(m, __shfl_xor(m, o, 32));
    float e = expf(v - m);
    float s = e;
    for (int o = 16; o > 0; o >>= 1) s += __shfl_xor(s, o, 32);
    float out = e / s;
    S[(size_t)row * 32 + lane] = out;
    float t = out;                               // faithful row sum of normalized S
    for (int o = 16; o > 0; o >>= 1) t += __shfl_xor(t, o, 32);
    if (lane == 0) rs[row] = t;
}

__global__ void ls_init_kernel(const float* __restrict__ S, float* __restrict__ LS, int total) {
    int i = blockIdx.x * blockDim.x + threadIdx.x;
    if (i < total) LS[i] = S[i];
}

// LS[row,k] += -(dinv2[row]*dinv2[col]) * S[col,k]   (8 edges per block)
__global__ void ls_scatter_kernel(const int* __restrict__ edges, const float* __restrict__ S,
                                  const float* __restrict__ dinv2, float* __restrict__ LS, int E) {
    int e = blockIdx.x * 8 + (threadIdx.x >> 5);
    int k = threadIdx.x & 31;
    if (e >= E) return;
    int r = edges[e];
    int c = edges[E + e];
    float w = -dinv2[r] * dinv2[c];
    if (w != 0.0f) atomicAdd(&LS[(size_t)r * 32 + k], w * S[(size_t)c * 32 + k]);
}

// ---------------------------------------------------------------------------
// new_adj[a,b] = sum_n S[n,a]*LS[n,b]  (block partial sums -> atomic finish)
// ---------------------------------------------------------------------------
__global__ void newadj_kernel(const float* __restrict__ S, const float* __restrict__ LS,
                              float* __restrict__ na, int N) {
    __shared__ float sS[8][32];
    __shared__ float sL[8][32];
    float acc[4] = {0.f, 0.f, 0.f, 0.f};
    int t = threadIdx.x;
    int r = t >> 5, c = t & 31;
    for (int base = blockIdx.x * 8; base < N; base += gridDim.x * 8) {
        int nrows = min(8, N - base);
        if (r < nrows) {
            sS[r][c] = S[(size_t)(base + r) * 32 + c];
            sL[r][c] = LS[(size_t)(base + r) * 32 + c];
        }
        __syncthreads();
        for (int i = 0; i < nrows; ++i) {
#pragma unroll
            for (int p = 0; p < 4; ++p) {
                int idx = t * 4 + p;
                acc[p] += sS[i][idx >> 5] * sL[i][idx & 31];
            }
        }
        __syncthreads();
    }
#pragma unroll
    for (int p = 0; p < 4; ++p) atomicAdd(&na[t * 4 + p], acc[p]);
}

// graph_embedding accumulator: geacc[d] = sum_n rs[n]*H[n,d]
// (mean_k (S^T H)[k,d] == (1/K) * sum_n rowsum(S)_n * H[n,d])
__global__ void ge_kernel(const float* __restrict__ H, const float* __restrict__ rs,
                          float* __restrict__ geacc, int N) {
    int d = threadIdx.x;                         // 256
    float acc = 0.0f;
    for (int n = blockIdx.x; n < N; n += gridDim.x)
        acc += rs[n] * H[(size_t)n * 256 + d];
    atomicAdd(&geacc[d], acc);
}

// ---------------------------------------------------------------------------
// Finalize: out[0..255] = geacc/32 ; out[256] = mean((norm_diag[j]-I[i,j])^2)
// ---------------------------------------------------------------------------
__global__ void finalize_kernel(const float* __restrict__ na, const float* __restrict__ geacc,
                                float* __restrict__ out) {
    __shared__ float nd[32];
    __shared__ float pp[32];
    int t = threadIdx.x;                         // 256 threads
    out[t] = geacc[t] * (1.0f / 32.0f);
    if (t < 32) {
        float s = 0.0f;
        for (int j = 0; j < 32; ++j) s += fabsf(na[t * 32 + j]);
        nd[t] = na[t * 32 + t] / fmaxf(s, 1e-12f);
    }
    __syncthreads();
    if (t < 32) {
        float p = 0.0f;
        for (int j = 0; j < 32; ++j) {
            float d = nd[j] - ((t == j) ? 1.0f : 0.0f);
            p += d * d;
        }
        pp[t] = p;
    }
    __syncthreads();
    if (t == 0) {
        float s = 0.0f;
        for (int i = 0; i < 32; ++i) s += pp[i];
        out[256] = s * (1.0f / 1024.0f);
    }
}

// ---------------------------------------------------------------------------
extern "C" void kernel_launch(void* const* d_in, const int* in_sizes, int n_in,
                              void* d_out, int out_size, void* d_ws, size_t ws_size,
                              hipStream_t stream) {
    const float* features = (const float*)d_in[0];   // [N,128]
    const int*   edges    = (const int*)d_in[1];     // [2,E]
    const float* W1       = (const float*)d_in[2];   // [128,256]
    const float* b1       = (const float*)d_in[3];   // [256]
    const float* fc1_W    = (const float*)d_in[4];   // [256,128]
    const float* fc1_b    = (const float*)d_in[5];   // [128]
    const float* fc2_W    = (const float*)d_in[6];   // [128,32]
    const float* fc2_b    = (const float*)d_in[7];   // [32]
    float* out = (float*)d_out;

    const int N = N_NODES, E = N_EDGES;

    // Workspace layout (floats)
    float* ws    = (float*)d_ws;
    float* XW    = ws;                       // 50000*256 = 12,800,000
    float* H     = XW + 12800000;            // 12,800,000
    float* AB    = H + 12800000;             //  6,400,000
    float* S     = AB + 6400000;             //  1,600,000  (logits, softmax in place)
    float* LS    = S + 1600000;              //  1,600,000
    float* deg1  = LS + 1600000;             //     50,000
    float* deg2  = deg1 + 50000;             //     50,000
    float* dinv1 = deg2 + 50000;             //     50,000
    float* dinv2 = dinv1 + 50000;            //     50,000
    float* rs    = dinv2 + 50000;            //     50,000
    float* na    = rs + 50000;               //      1,024
    float* geacc = na + 1024;                //        256

    // Zero the accumulators (graph-capture safe)
    hipMemsetAsync(deg1, 0, 2 * 50000 * sizeof(float), stream);   // deg1+deg2
    hipMemsetAsync(na, 0, (1024 + 256) * sizeof(float), stream);  // na+geacc

    // Degrees + inverse sqrt
    deg_kernel<<<(E + 255) / 256, 256, 0, stream>>>(edges, deg1, deg2, E);
    dinv_kernel<<<(N + 255) / 256, 256, 0, stream>>>(deg1, deg2, dinv1, dinv2, N);

    // XW = features @ W1        M=50000 N=256 K=128 -> 50000 waves
    gemm_wmma_f32<0><<<6250, 256, 0, stream>>>(features, W1, nullptr, XW, N, 256, 128);

    // H = b1 + selfloop + edge scatter
    h_init_kernel<<<N, 256, 0, stream>>>(XW, b1, dinv1, H);
    h_scatter_kernel<<<E, 256, 0, stream>>>(edges, XW, dinv1, H, E);

    // abstract = tanh(H @ fc1_W + fc1_b)   M=50000 N=128 K=256 -> 25000 waves
    gemm_wmma_f32<1><<<3125, 256, 0, stream>>>(H, fc1_W, fc1_b, AB, N, 128, 256);

    // logits = abstract @ fc2_W + fc2_b    M=50000 N=32 K=128 -> 6250 waves
    gemm_wmma_f32<0><<<782, 256, 0, stream>>>(AB, fc2_W, fc2_b, S, N, 32, 128);

    // S = softmax(logits) (+ row sums), LS = S + Laplacian scatter
    softmax_kernel<<<(N * 32 + 255) / 256, 256, 0, stream>>>(S, rs, N);
    ls_init_kernel<<<(N * 32 + 255) / 256, 256, 0, stream>>>(S, LS, N * 32);
    ls_scatter_kernel<<<(E + 7) / 8, 256, 0, stream>>>(edges, S, dinv2, LS, E);

    // new_adj = S^T LS ; embedding column-sum ; finalize
    newadj_kernel<<<512, 256, 0, stream>>>(S, LS, na, N);
    ge_kernel<<<1024, 256, 0, stream>>>(H, rs, geacc, N);
    finalize_kernel<<<1, 256, 0, stream>>>(na, geacc, out);

    (void)in_sizes; (void)n_in; (void)out_size; (void)ws_size;
}